// GCN_12979391168796
// MI455X (gfx1250) — compile-verified
//
#include <hip/hip_runtime.h>
#include <hip/hip_bf16.h>

typedef __attribute__((ext_vector_type(2))) float v2f;
typedef __attribute__((ext_vector_type(8))) float v8f;

#define N_NODES 100000
#define KDIM    128   // feature dim into both GEMMs

// Guaranteed-native non-returning f32 atomic add (L2 scatter path).
__device__ __forceinline__ void atomic_add_f32(float* p, float v) {
  asm volatile("global_atomic_add_f32 %0, %1, off" : : "v"(p), "v"(v) : "memory");
}

// ---------------------------------------------------------------------------
// SpMM scatter: out[r,:] += val * x[c,:].  One edge per wave32; each lane
// gathers float4 (16B) of x and issues 4 global_atomic_add_f32 (L2-resident).
// ---------------------------------------------------------------------------
__global__ __launch_bounds__(256) void spmm_edge_atomic(
    const int* __restrict__ rows, const int* __restrict__ cols,
    const float* __restrict__ vals, const float* __restrict__ x,
    float* __restrict__ out, int E) {
  int e = blockIdx.x * 8 + (threadIdx.x >> 5);
  if (e >= E) return;
  int lane = threadIdx.x & 31;
  int r = rows[e];
  int c = cols[e];
  float v = vals[e];
  float4 xv = *((const float4*)(x + (size_t)c * 128) + lane);
  float* o = out + (size_t)r * 128 + lane * 4;
  atomic_add_f32(o + 0, xv.x * v);
  atomic_add_f32(o + 1, xv.y * v);
  atomic_add_f32(o + 2, xv.z * v);
  atomic_add_f32(o + 3, xv.w * v);
}

// ---------------------------------------------------------------------------
// Dense GEMM  C[M,N] = relu?(A[M,128] @ B[128,N] + bias[N])  via
// V_WMMA_F32_16X16X4_F32.  8 waves/block, one 16xNCOLS C-strip (NCOLS/16 tiles) per wave.
//
// B is staged in LDS K-pair interleaved: element (k,n) lives at
//   Bs[(k>>1)*LDSP + n*2 + (k&1)]
// so a lane's B fragment {B[kb][n], B[kb+1][n]} is a single aligned
// ds_load_b64.  LDSP chosen so the two half-waves (k-pair 2ks vs 2ks+1)
// land in disjoint 32-bank windows (LDSP % 64 floats == 32).
// A tile (16x128) is preloaded into VGPRs and reused across column tiles.
// ---------------------------------------------------------------------------
template <int NCOLS, bool RELU>
__global__ __launch_bounds__(256) void gemm_wmma_f32(
    const float* __restrict__ A, const float* __restrict__ B,
    const float* __restrict__ bias, float* __restrict__ C, int M) {
  constexpr int K = KDIM;
  constexpr int LDSP = 2 * NCOLS + 32;      // floats per k-pair row (padded)
  __shared__ __align__(16) float Bs[(K / 2) * LDSP];

  // Cooperative load of B[128 x NCOLS] into interleaved LDS layout.
  for (int i = threadIdx.x; i < K * NCOLS; i += 256) {
    int k = i / NCOLS, n = i % NCOLS;
    Bs[(k >> 1) * LDSP + n * 2 + (k & 1)] = B[i];
  }
  __syncthreads();

  const int wave = threadIdx.x >> 5;
  const int lane = threadIdx.x & 31;
  const int tileRow = blockIdx.x * 8 + wave;     // 16-row tile index
  if (tileRow * 16 >= M) return;                 // M % 16 == 0 (100000/16=6250)
  const int row0 = tileRow * 16;

  const int mlane = lane & 15;                   // M (for A) / N (for B,C)
  const int khalf = (lane >> 4) * 2;             // K offset: lanes 16-31 -> +2

  // Preload A fragments: 32 k-steps of 16x4, 2 floats per lane each.
  v2f afrag[K / 4];
  const float* arow = A + (size_t)(row0 + mlane) * K + khalf;
#pragma unroll
  for (int ks = 0; ks < K / 4; ++ks)
    afrag[ks] = *(const v2f*)(arow + ks * 4);    // global_load_b64

#pragma unroll
  for (int nt = 0; nt < NCOLS / 16; ++nt) {
    const int n = nt * 16 + mlane;
    const float bv = bias[n];
    v8f acc;
#pragma unroll
    for (int i = 0; i < 8; ++i) acc[i] = bv;     // fold bias into accumulator

#pragma unroll
    for (int ks = 0; ks < K / 4; ++ks) {
      // kb = ks*4 + khalf (even); k-pair index = kb>>1 = ks*2 + (lane>>4)
      const int kpair = ks * 2 + (lane >> 4);
      v2f bfrag = *(const v2f*)&Bs[kpair * LDSP + n * 2];   // ds_load_b64
      acc = __builtin_amdgcn_wmma_f32_16x16x4_f32(
          /*neg_a=*/false, afrag[ks], /*neg_b=*/false, bfrag,
          /*c_mod=*/(short)0, acc, /*reuse_a=*/false, /*reuse_b=*/false);
    }

    // C/D layout: VGPR v -> M = v + 8*(lane>=16), N = lane&15 (+ tile offset)
    const int mbase = row0 + ((lane >> 4) * 8);
#pragma unroll
    for (int v = 0; v < 8; ++v) {
      float val = acc[v];
      if (RELU) val = fmaxf(val, 0.0f);
      C[(size_t)(mbase + v) * NCOLS + n] = val;
    }
  }
}

// ---------------------------------------------------------------------------
// Pipeline: zero h1 -> SpMM(x)->h1 -> GEMM1+ReLU->h2 -> zero h1 ->
//           SpMM(h2)->h1 -> GEMM2->out
// ---------------------------------------------------------------------------
extern "C" void kernel_launch(void* const* d_in, const int* in_sizes, int n_in,
                              void* d_out, int out_size, void* d_ws, size_t ws_size,
                              hipStream_t stream) {
  const float* x  = (const float*)d_in[0];
  const int*   er = (const int*)d_in[1];
  const int*   ec = (const int*)d_in[2];
  const float* ev = (const float*)d_in[3];
  const float* W1 = (const float*)d_in[4];
  const float* b1 = (const float*)d_in[5];
  const float* W2 = (const float*)d_in[6];
  const float* b2 = (const float*)d_in[7];
  float* out = (float*)d_out;
  const int E = in_sizes[1];

  const size_t hElems = (size_t)N_NODES * 128;
  float* h1 = (float*)d_ws;           // 51.2 MB
  float* h2 = h1 + hElems;            // 51.2 MB

  const int spmmBlocks = (E + 7) / 8;                // 1 edge per wave
  const int gemmBlocks = ((N_NODES / 16) + 7) / 8;   // 6250 tiles -> 782 blocks

  hipMemsetAsync(h1, 0, hElems * sizeof(float), stream);
  spmm_edge_atomic<<<spmmBlocks, 256, 0, stream>>>(er, ec, ev, x, h1, E);
  gemm_wmma_f32<128, true><<<gemmBlocks, 256, 0, stream>>>(h1, W1, b1, h2, N_NODES);
  hipMemsetAsync(h1, 0, hElems * sizeof(float), stream);
  spmm_edge_atomic<<<spmmBlocks, 256, 0, stream>>>(er, ec, ev, h2, h1, E);
  gemm_wmma_f32<64, false><<<gemmBlocks, 256, 0, stream>>>(h1, W2, b2, out, N_NODES);
}